// ProcessContinuous_11227044512450
// MI455X (gfx1250) — compile-verified
//
#include <hip/hip_runtime.h>
#include <stdint.h>

typedef __attribute__((ext_vector_type(4))) float f32x4;
typedef __attribute__((ext_vector_type(4))) int   v4i;

#define B_DIM 32
#define M_DIM 2048
#define E_DIM 128
// fused row length = E*5 = 640 floats per (m,b) pair

#if defined(__gfx1250__) && __has_builtin(__builtin_amdgcn_global_load_async_to_lds_b128)
#define USE_ASYNC_LDS 1
#else
#define USE_ASYNC_LDS 0
#endif

typedef __attribute__((address_space(1))) v4i* as1_v4i_p;
typedef __attribute__((address_space(3))) v4i* as3_v4i_p;

// Launch contract (enforced by kernel_launch):
//   stride = gridDim.x*256 chunks; stride*4 must be a multiple of 640*32
//   (=> per-iteration bm advance is a multiple of 32: j and b invariant),
//   and nChunks == iters * stride (exact cover, no tail).
__global__ __launch_bounds__(256) void fused_embed_scramble_kernel(
    const float* __restrict__ x,
    const float* __restrict__ w_bet,   const float* __restrict__ b_bet,
    const float* __restrict__ w_stack, const float* __restrict__ b_stack,
    const float* __restrict__ w_call,  const float* __restrict__ b_call,
    const float* __restrict__ w_odds,  const float* __restrict__ b_odds,
    const float* __restrict__ emb,     // [5,128]
    float* __restrict__ out, int nChunks, int iters)
{
    // raw tables: [0:512) = w[4][128] (bet,stack,call,odds)
    //             [512:1024) = b[4][128]
    //             [1024:1664) = emb[5][128]
    __shared__ float lds_raw[1664];
    __shared__ f32x4 lds_wtab[160];   // 640 fused weights
    __shared__ f32x4 lds_ctab[160];   // 640 fused bias+emb

    const int tid = threadIdx.x;

    const float* tabs[8] = { w_bet, w_stack, w_call, w_odds,
                             b_bet, b_stack, b_call, b_odds };

    // ---- Phase 1: stage 416 float4 slots (1664 dwords) into LDS ----
    for (int s = tid; s < 416; s += 256) {
        const float* src;
        int dst;
        if (s < 256) {              // 8 arrays x 32 float4 slots
            int arr = s >> 5;
            int o   = (s & 31) << 2;
            src = tabs[arr] + o;
            dst = arr * 128 + o;
        } else {                    // emb: 160 float4 slots
            int o = (s - 256) << 2;
            src = emb + o;
            dst = 1024 + o;
        }
#if USE_ASYNC_LDS
        __builtin_amdgcn_global_load_async_to_lds_b128(
            (as1_v4i_p)(v4i*)(void*)src,
            (as3_v4i_p)(v4i*)(void*)&lds_raw[dst],
            0, 0);
#else
        *(f32x4*)&lds_raw[dst] = *(const f32x4*)src;
#endif
    }
#if USE_ASYNC_LDS
#if __has_builtin(__builtin_amdgcn_s_wait_asynccnt)
    __builtin_amdgcn_s_wait_asynccnt(0);
#else
    asm volatile("s_wait_asynccnt 0" ::: "memory");
#endif
#endif
    __syncthreads();

    // ---- Phase 2: build fused tables w_tab[j], c_tab[j], j = e*5+f ----
    for (int j = tid; j < 640; j += 256) {
        int e = j / 5;
        int f = j - e * 5;
        // f: 0->bet(0), 1->stack(1), 2->stack(1), 3->call(2), 4->odds(3)
        int sel = (f == 0) ? 0 : ((f <= 2) ? 1 : (f - 1));
        float w = lds_raw[sel * 128 + e];
        float c = lds_raw[512 + sel * 128 + e] + lds_raw[1024 + f * 128 + e];
        ((float*)lds_wtab)[j] = w;
        ((float*)lds_ctab)[j] = c;
    }
    __syncthreads();

    // ---- Phase 3: stream the 168 MB output as aligned float4 chunks ----
    const int chunk0 = blockIdx.x * 256 + tid;
    if (chunk0 >= nChunks) return;

    const int stride   = gridDim.x * 256;      // chunks per iteration step
    const int bmStride = stride / 160;         // (stride*4)/640 rows per step
                                               // multiple of 32 by contract

    const int idx = chunk0 << 2;               // flat float index, < 2^26
    int bm        = idx / 640;                 // bm = m*32 + b (scrambled row)
    const int j   = idx - bm * 640;            // invariant across iterations

    // Hoisted per-thread constants
    const f32x4 w4 = lds_wtab[j >> 2];
    const f32x4 c4 = lds_ctab[j >> 2];
    int f0 = j % 5;
    int f1 = (f0 + 1 == 5) ? 0 : f0 + 1;
    int f2 = (f1 + 1 == 5) ? 0 : f1 + 1;
    int f3 = (f2 + 1 == 5) ? 0 : f2 + 1;

    const int b = bm & 31;                     // invariant (bmStride % 32 == 0)
    int m       = bm >> 5;                     // advances by bmStride/32

    const float* xb   = x + (b * M_DIM + m) * 5;   // x[b][m][0..4]
    const int    xAdv = (bmStride >> 5) * 5;       // floats per iteration
    f32x4*       op   = (f32x4*)out + chunk0;

    #pragma unroll 4
    for (int it = 0; it < iters; ++it) {
        float x0 = xb[f0];
        float x1 = xb[f1];
        float x2 = xb[f2];
        float x3 = xb[f3];

        f32x4 o;
        o.x = __builtin_fmaf(x0, w4.x, c4.x);
        o.y = __builtin_fmaf(x1, w4.y, c4.y);
        o.z = __builtin_fmaf(x2, w4.z, c4.z);
        o.w = __builtin_fmaf(x3, w4.w, c4.w);

        __builtin_nontemporal_store(o, op);

        xb += xAdv;
        op += stride;
    }
}

extern "C" void kernel_launch(void* const* d_in, const int* in_sizes, int n_in,
                              void* d_out, int out_size, void* d_ws, size_t ws_size,
                              hipStream_t stream) {
    (void)in_sizes; (void)n_in; (void)d_ws; (void)ws_size;

    const float* x       = (const float*)d_in[0];
    const float* w_bet   = (const float*)d_in[1];
    const float* b_bet   = (const float*)d_in[2];
    const float* w_stack = (const float*)d_in[3];
    const float* b_stack = (const float*)d_in[4];
    const float* w_call  = (const float*)d_in[5];
    const float* b_call  = (const float*)d_in[6];
    const float* w_odds  = (const float*)d_in[7];
    const float* b_odds  = (const float*)d_in[8];
    const float* emb     = (const float*)d_in[9];
    float* out = (float*)d_out;

    int nChunks = out_size >> 2;           // 10,485,760 float4 chunks

    // stride*4 = 2,621,440 floats = 4096 * 640  (4096 % 32 == 0)  ✓
    // nChunks  = 16 * stride                                      ✓
    dim3 grid(2560), block(256);
    int stride = 2560 * 256;
    int iters  = nChunks / stride;         // 16, exact

    hipLaunchKernelGGL(fused_embed_scramble_kernel, grid, block, 0, stream,
                       x, w_bet, b_bet, w_stack, b_stack, w_call, b_call,
                       w_odds, b_odds, emb, out, nChunks, iters);
}